// RuchbahExpertOrientedRouter_4131758538904
// MI455X (gfx1250) — compile-verified
//
#include <hip/hip_runtime.h>
#include <math.h>

typedef __attribute__((ext_vector_type(2))) float v2f;
typedef __attribute__((ext_vector_type(4))) float v4f;
typedef __attribute__((ext_vector_type(8))) float v8f;

#define HDIM 2048
#define NCOLS 128
#define KTILE 64
#define MTILE 64
#define LDK 68                      // LDS row stride (floats) for staging tiles
#define LDC 132                     // LDS row stride (floats) for C/epilogue tile
#define BUFSZ (64 * LDK + 128 * LDK) // 13056 floats per staging buffer
#define NCHUNK (HDIM / KTILE)       // 32

// ---------------------------------------------------------------------------
// Kernel 1 (1 block, 64 threads): fold the expert side into W3[64][64], c3[64]
//   ee = silu(exp_emb @ ec_w1.T + ec_b1) @ ec_w2.T + ec_b2        [64,64]
//   W2 = bil_w @ ee.T                                             [64,64]
//   W3[d,e] = sum_d' ie_w2[d',d] * W2[d',e]
//   c3[e]   = sum_d' ie_b2[d']   * W2[d',e] + bil_b
// ws layout (floats): [0..4095] = W3 row-major, [4096..4159] = c3
// ---------------------------------------------------------------------------
__global__ void ruchbah_setup_kernel(
    const float* __restrict__ exp_emb,
    const float* __restrict__ ec_w1, const float* __restrict__ ec_b1,
    const float* __restrict__ ec_w2, const float* __restrict__ ec_b2,
    const float* __restrict__ ie_w2, const float* __restrict__ ie_b2,
    const float* __restrict__ bil_w, const float* __restrict__ bil_b,
    float* __restrict__ ws)
{
    __shared__ float slds[64 * 65];
    __shared__ float eelds[64 * 65];
    __shared__ float w2lds[64 * 65];
    const int t = threadIdx.x;  // 0..63

    for (int j = 0; j < 64; ++j) {
        float acc = ec_b1[j];
        for (int d = 0; d < 64; ++d) acc += exp_emb[t * 64 + d] * ec_w1[j * 64 + d];
        slds[t * 65 + j] = acc / (1.0f + expf(-acc));
    }
    for (int j = 0; j < 64; ++j) {
        float acc = ec_b2[j];
        for (int d = 0; d < 64; ++d) acc += slds[t * 65 + d] * ec_w2[j * 64 + d];
        eelds[t * 65 + j] = acc;
    }
    __syncthreads();
    for (int e = 0; e < 64; ++e) {
        float acc = 0.0f;
        for (int d = 0; d < 64; ++d) acc += bil_w[t * 64 + d] * eelds[e * 65 + d];
        w2lds[t * 65 + e] = acc;
    }
    __syncthreads();
    for (int e = 0; e < 64; ++e) {
        float acc = 0.0f;
        for (int d = 0; d < 64; ++d) acc += ie_w2[d * 64 + t] * w2lds[d * 65 + e];
        ws[t * 64 + e] = acc;
    }
    {
        float acc = bil_b[0];
        for (int d = 0; d < 64; ++d) acc += ie_b2[d] * w2lds[d * 65 + t];
        ws[4096 + t] = acc;
    }
}

// ---------------------------------------------------------------------------
// Kernel 2: fused [T x 2048] x [2048 x 128] fp32 WMMA GEMM (gate logits ||
// IE hidden pre-activation). Double-buffered LDS staging with register
// prefetch: global loads for chunk i+1 issue before the WMMA loop on chunk i,
// one barrier per chunk. Epilogue fully in LDS: SiLU, folded bilinear,
// softmax over 64 experts, top-2 + renormalized top-2 softmax.
// out layout (floats): [0,32768) top_scores, [32768,65536) top_idx (int bits),
//                      [65536, 65536+16384*64) scores
// ---------------------------------------------------------------------------
__global__ __launch_bounds__(256)
void ruchbah_router_kernel(
    const float* __restrict__ x,
    const float* __restrict__ gate_w,
    const float* __restrict__ ie_w1,
    const float* __restrict__ ie_b1,
    const float* __restrict__ ws,     // W3 + c3 from setup kernel
    float* __restrict__ out)
{
    __shared__ float lds[2 * BUFSZ];  // 104448 bytes

    const int tid  = threadIdx.x;
    const int lane = tid & 31;
    const int wid  = tid >> 5;
    const int wm   = wid & 1;   // M quadrant (rows 0/32)
    const int wn   = wid >> 1;  // N quadrant (cols 0/32/64/96)
    const int mbase = blockIdx.x * MTILE;

    const int halfk = (lane >> 4) * 2;           // 0 or 2
    const int mrow0 = wm * 32 + (lane & 15);
    const int ncol0 = wn * 32 + (lane & 15);

    v8f c[2][2] = {};
    v4f xr[4];
    v4f wr[8];

    auto stage_load = [&](int kc) {
        #pragma unroll
        for (int j = 0; j < 4; ++j) {
            int v = tid + 256 * j;  int row = v >> 4;  int k4 = v & 15;
            xr[j] = *(const v4f*)(x + (size_t)(mbase + row) * HDIM + kc + k4 * 4);
        }
        #pragma unroll
        for (int j = 0; j < 8; ++j) {
            int v = tid + 256 * j;  int n = v >> 4;  int k4 = v & 15;
            const float* src = (n < 64) ? (gate_w + (size_t)n * HDIM)
                                        : (ie_w1 + (size_t)(n - 64) * HDIM);
            wr[j] = *(const v4f*)(src + kc + k4 * 4);
        }
    };
    auto stage_store = [&](float* buf) {
        float* xsb = buf;
        float* wsb = buf + 64 * LDK;
        #pragma unroll
        for (int j = 0; j < 4; ++j) {
            int v = tid + 256 * j;  int row = v >> 4;  int k4 = v & 15;
            *(v4f*)(xsb + row * LDK + k4 * 4) = xr[j];
        }
        #pragma unroll
        for (int j = 0; j < 8; ++j) {
            int v = tid + 256 * j;  int n = v >> 4;  int k4 = v & 15;
            *(v4f*)(wsb + n * LDK + k4 * 4) = wr[j];
        }
    };

    // prologue: stage chunk 0 into buffer 0
    stage_load(0);
    stage_store(lds);
    __syncthreads();

    for (int i = 0; i < NCHUNK; ++i) {
        float* cur = lds + (i & 1) * BUFSZ;
        float* nxt = lds + ((i + 1) & 1) * BUFSZ;
        const bool more = (i + 1 < NCHUNK);

        if (more) stage_load((i + 1) * KTILE);   // overlaps with WMMA below

        float* xs  = cur;
        float* wsh = cur + 64 * LDK;
        #pragma unroll
        for (int ks = 0; ks < KTILE; ks += 4) {
            v2f a0 = *(const v2f*)(xs  + (mrow0     ) * LDK + ks + halfk);
            v2f a1 = *(const v2f*)(xs  + (mrow0 + 16) * LDK + ks + halfk);
            v2f b0 = *(const v2f*)(wsh + (ncol0     ) * LDK + ks + halfk);
            v2f b1 = *(const v2f*)(wsh + (ncol0 + 16) * LDK + ks + halfk);
            c[0][0] = __builtin_amdgcn_wmma_f32_16x16x4_f32(false, a0, false, b0, (short)0, c[0][0], false, false);
            c[0][1] = __builtin_amdgcn_wmma_f32_16x16x4_f32(false, a0, false, b1, (short)0, c[0][1], false, false);
            c[1][0] = __builtin_amdgcn_wmma_f32_16x16x4_f32(false, a1, false, b0, (short)0, c[1][0], false, false);
            c[1][1] = __builtin_amdgcn_wmma_f32_16x16x4_f32(false, a1, false, b1, (short)0, c[1][1], false, false);
        }

        // nxt buffer's previous readers finished before this iteration's start
        if (more) stage_store(nxt);
        __syncthreads();
    }

    // ---- epilogue: dump C tile + stage W3/c3 into buffer-0 region ----
    float* ct = lds + 64 * LDK;   // [64][LDC] = 8448 floats, fits in 128*LDK
    {
        const int m0   = wm * 32;
        const int n0   = wn * 32 + (lane & 15);
        const int rowh = (lane >> 4) * 8;
        #pragma unroll
        for (int mi = 0; mi < 2; ++mi)
            #pragma unroll
            for (int ni = 0; ni < 2; ++ni)
                #pragma unroll
                for (int r = 0; r < 8; ++r)
                    ct[(m0 + 16 * mi + rowh + r) * LDC + n0 + 16 * ni] = c[mi][ni][r];
    }
    float* w3 = lds;            // [64*64]
    float* c3 = lds + 4096;     // [64]
    for (int i = tid; i < 4096 + 64; i += 256) lds[i] = ws[i];
    __syncthreads();

    if (tid < 64) {
        const int t     = tid;
        const int token = mbase + t;
        // h = silu(u + ie_b1), in place in columns 64..127
        for (int d = 0; d < 64; ++d) {
            float u = ct[t * LDC + 64 + d] + ie_b1[d];
            ct[t * LDC + 64 + d] = u / (1.0f + expf(-u));
        }
        // combined = logits + 0.3 * (h @ W3 + c3), in place in columns 0..63
        float mx = -3.0e38f;
        for (int e = 0; e < 64; ++e) {
            float acc = c3[e];
            for (int d = 0; d < 64; ++d)
                acc += ct[t * LDC + 64 + d] * w3[d * 64 + e];
            float comb = ct[t * LDC + e] + 0.3f * acc;
            ct[t * LDC + e] = comb;
            mx = fmaxf(mx, comb);
        }
        float sum = 0.0f;
        for (int e = 0; e < 64; ++e) sum += expf(ct[t * LDC + e] - mx);
        const float inv = 1.0f / sum;

        float v1 = -1.0f, v2 = -1.0f;
        int   i1 = 0,     i2 = 0;
        float* scores = out + 65536;
        for (int e = 0; e < 64; ++e) {
            float s = expf(ct[t * LDC + e] - mx) * inv;
            scores[(size_t)token * 64 + e] = s;
            if (s > v1)      { v2 = v1; i2 = i1; v1 = s; i1 = e; }
            else if (s > v2) { v2 = s;  i2 = e; }
        }
        // softmax over the two top probabilities
        float e2 = expf(v2 - v1);
        float p1 = 1.0f / (1.0f + e2);
        out[token * 2 + 0] = p1;
        out[token * 2 + 1] = e2 * p1;
        int* ti = (int*)(out + 32768);
        ti[token * 2 + 0] = i1;
        ti[token * 2 + 1] = i2;
    }
}

extern "C" void kernel_launch(void* const* d_in, const int* in_sizes, int n_in,
                              void* d_out, int out_size, void* d_ws, size_t ws_size,
                              hipStream_t stream) {
    const float* x       = (const float*)d_in[0];
    const float* gate_w  = (const float*)d_in[1];
    const float* exp_emb = (const float*)d_in[2];
    const float* ec_w1   = (const float*)d_in[3];
    const float* ec_b1   = (const float*)d_in[4];
    const float* ec_w2   = (const float*)d_in[5];
    const float* ec_b2   = (const float*)d_in[6];
    const float* ie_w1   = (const float*)d_in[7];
    const float* ie_b1   = (const float*)d_in[8];
    const float* ie_w2   = (const float*)d_in[9];
    const float* ie_b2   = (const float*)d_in[10];
    const float* bil_w   = (const float*)d_in[11];
    const float* bil_b   = (const float*)d_in[12];
    float* ws  = (float*)d_ws;
    float* out = (float*)d_out;

    ruchbah_setup_kernel<<<1, 64, 0, stream>>>(exp_emb, ec_w1, ec_b1, ec_w2, ec_b2,
                                               ie_w2, ie_b2, bil_w, bil_b, ws);
    ruchbah_router_kernel<<<256, 256, 0, stream>>>(x, gate_w, ie_w1, ie_b1, ws, out);
}